// DimeNet_11012296147742
// MI455X (gfx1250) — compile-verified
//
#include <hip/hip_runtime.h>
#include <math.h>

// ---------------------------------------------------------------------------
// DimeNet forward, MI455X (gfx1250, wave32).
// Core engine: fp32 WMMA GEMM (V_WMMA_F32_16X16X4_F32), N fixed at 128.
// A and B tiles staged in LDS (branch-free inner loop); A staging uses
// gfx1250 async global->LDS (ASYNCcnt) on the in-bounds fast path.
// Bilinear interaction restructured from triplet-side (604 GFLOP) to
// edge-side (per-edge G accumulation + one [E,1024]@[1024,128] GEMM).
// ---------------------------------------------------------------------------

typedef float v2f __attribute__((ext_vector_type(2)));
typedef float v8f __attribute__((ext_vector_type(8)));

#define PM_NONE        0
#define PM_SILU        1
#define PM_SILU_RBMUL  2
#define PM_ADD         3
#define PM_SILU_ADD    4

#define KC    32      // K chunk per LDS stage
#define TSTR  36      // 32 + 4 pad dwords -> conflict-free ds_load_b64 frags

#if __has_builtin(__builtin_amdgcn_global_load_async_to_lds_b128) && \
    __has_builtin(__builtin_amdgcn_s_wait_asynccnt)
#define HAVE_ASYNC_LDS 1
typedef int v4i_vs __attribute__((vector_size(16)));
typedef __attribute__((address_space(1))) v4i_vs g_v4i;   // global int4
typedef __attribute__((address_space(3))) v4i_vs l_v4i;   // LDS int4
#else
#define HAVE_ASYNC_LDS 0
#endif

__device__ __forceinline__ float silu_f(float v) {
    return v / (1.f + __expf(-v));
}

// C[M,128] = post(A[M,K] @ B[K,128] + bias). 8 waves, each owns a 16x128 strip.
__global__ __launch_bounds__(256) void wmma_gemm_n128(
    const float* __restrict__ A, int lda,
    const float* __restrict__ B,        // [K,128] row-major
    const float* __restrict__ bias,     // [128] or null
    const float* __restrict__ post,     // [M,128] for ADD / SILU_ADD
    const float* __restrict__ rbf6,     // [M,6]   for SILU_RBMUL
    const float* __restrict__ Wr,       // [6,128] for SILU_RBMUL
    float* __restrict__ C,
    int M, int K, int postmode)
{
    __shared__ float As[128 * TSTR];    // A chunk: As[row][k]
    __shared__ float Bst[128 * TSTR];   // B chunk transposed: Bst[n][k]
    __shared__ float WrS[6 * 128];

    const int tid  = threadIdx.x;
    const int lane = tid & 31;
    const int wave = tid >> 5;
    const int m0   = blockIdx.x * 128;

    if (postmode == PM_SILU_RBMUL)
        for (int i = tid; i < 6 * 128; i += 256) WrS[i] = Wr[i];

    v8f acc[8] = {};

    const int  mloc  = 16 * wave + (lane & 15);
    const int  kaoff = 2 * (lane >> 4);     // lanes 16..31 hold K+2,K+3
    const bool fullM = (m0 + 128 <= M);

    for (int k0 = 0; k0 < K; k0 += KC) {
        __syncthreads();
        // ---- stage B chunk transposed (coalesced float4 reads) ----
        for (int kk2 = tid >> 5; kk2 < KC; kk2 += 8) {
            int n4 = lane * 4;
            int kg = k0 + kk2;
            float4 v = make_float4(0.f, 0.f, 0.f, 0.f);
            if (kg < K) v = *(const float4*)(B + (size_t)kg * 128 + n4);
            Bst[(n4 + 0) * TSTR + kk2] = v.x;
            Bst[(n4 + 1) * TSTR + kk2] = v.y;
            Bst[(n4 + 2) * TSTR + kk2] = v.z;
            Bst[(n4 + 3) * TSTR + kk2] = v.w;
        }
        // ---- stage A chunk ----
        const bool fastA = fullM && (k0 + KC <= K);
#if HAVE_ASYNC_LDS
        if (fastA) {
            // per-lane global->LDS async B128 copies (ASYNCcnt)
            #pragma unroll
            for (int it = 0; it < 4; it++) {
                int idx = it * 256 + tid;        // 0..1023
                int r = idx >> 3, q = idx & 7;   // row, float4-quad
                const float* gp = A + (size_t)(m0 + r) * lda + k0 + q * 4;
                __builtin_amdgcn_global_load_async_to_lds_b128(
                    (g_v4i*)gp, (l_v4i*)&As[r * TSTR + q * 4], 0, 0);
            }
            __builtin_amdgcn_s_wait_asynccnt(0);
        } else
#endif
        {
            for (int it = 0; it < 16; it++) {
                int idx = it * 256 + tid;        // 0..4095
                int r = idx >> 5, kk2 = idx & 31;
                int gr = m0 + r, gk = k0 + kk2;
                float v = 0.f;
                if (fastA || (gr < M && gk < K)) v = A[(size_t)gr * lda + gk];
                As[r * TSTR + kk2] = v;
            }
        }
        __syncthreads();

        int kc = K - k0; if (kc > KC) kc = KC;
        int kc4 = (kc + 3) & ~3;
        #pragma unroll 2
        for (int kk = 0; kk < kc4; kk += 4) {
            v2f afr;
            afr.x = As[mloc * TSTR + kk + kaoff];
            afr.y = As[mloc * TSTR + kk + kaoff + 1];
            v2f bfr[8];
            #pragma unroll
            for (int nt = 0; nt < 8; nt++) {
                int coln = nt * 16 + (lane & 15);
                bfr[nt].x = Bst[coln * TSTR + kk + kaoff];
                bfr[nt].y = Bst[coln * TSTR + kk + kaoff + 1];
            }
            #pragma unroll
            for (int nt = 0; nt < 8; nt++)
                acc[nt] = __builtin_amdgcn_wmma_f32_16x16x4_f32(
                    false, afr, false, bfr[nt], (short)0, acc[nt], false, false);
        }
    }

    // Epilogue. C/D layout: VGPR r -> (M = r + 8*(lane>>4), N = lane&15).
    #pragma unroll
    for (int nt = 0; nt < 8; nt++) {
        #pragma unroll
        for (int r = 0; r < 8; r++) {
            int grow = m0 + 16 * wave + r + 8 * (lane >> 4);
            int col  = nt * 16 + (lane & 15);
            if (grow < M) {
                float v = acc[nt][r];
                if (bias) v += bias[col];
                if (postmode == PM_SILU || postmode == PM_SILU_RBMUL ||
                    postmode == PM_SILU_ADD)
                    v = silu_f(v);
                if (postmode == PM_SILU_RBMUL) {
                    float rb = 0.f;
                    #pragma unroll
                    for (int q = 0; q < 6; q++)
                        rb += rbf6[(size_t)grow * 6 + q] * WrS[q * 128 + col];
                    v *= rb;
                }
                if (postmode == PM_ADD || postmode == PM_SILU_ADD)
                    v += post[(size_t)grow * 128 + col];
                C[(size_t)grow * 128 + col] = v;
            }
        }
    }
}

// ------------------------- geometry kernels --------------------------------

__global__ __launch_bounds__(256) void edge_geom_kernel(
    const float* __restrict__ pos, const int* __restrict__ ei, int E,
    const float* __restrict__ zeros, const float* __restrict__ norms,
    float* __restrict__ rbf, float* __restrict__ rade)
{
    int e = blockIdx.x * 256 + threadIdx.x;
    if (e >= E) return;
    int j = ei[e], i = ei[E + e];           // j=row, i=col
    float dx = pos[i * 3 + 0] - pos[j * 3 + 0];
    float dy = pos[i * 3 + 1] - pos[j * 3 + 1];
    float dz = pos[i * 3 + 2] - pos[j * 3 + 2];
    float dist = sqrtf(dx * dx + dy * dy + dz * dz);
    float xd = dist * 0.2f;                  // /CUTOFF
    float env = 0.f;
    if (xd < 1.f) {
        float x2 = xd * xd, x4 = x2 * x2, x5 = x4 * xd;
        env = 1.f / xd - 28.f * x5 + 48.f * x5 * xd - 21.f * x5 * x2;
    }
    const float PI = 3.14159265358979f;
    #pragma unroll
    for (int r = 0; r < 6; r++)
        rbf[(size_t)e * 6 + r] = env * __sinf((float)(r + 1) * PI * xd);
    for (int l = 0; l < 7; l++) {
        for (int r = 0; r < 6; r++) {
            float z = zeros[l * 6 + r] * xd;
            float s, c;
            __sincosf(z, &s, &c);
            float j0 = s / z;
            float jl = j0;
            if (l > 0) {
                float j1 = s / (z * z) - c / z;
                float jm = j0, jc = j1;
                for (int ll = 2; ll <= l; ll++) {
                    float tn = (float)(2 * ll - 1) / z * jc - jm;
                    jm = jc; jc = tn;
                }
                jl = jc;
            }
            rade[(size_t)e * 42 + l * 6 + r] = env * norms[l * 6 + r] * jl;
        }
    }
}

__global__ __launch_bounds__(256) void trip_geom_kernel(
    const float* __restrict__ pos, const int* __restrict__ ii,
    const int* __restrict__ ij, const int* __restrict__ ik,
    float* __restrict__ cbf, int T)
{
    int t = blockIdx.x * 256 + threadIdx.x;
    if (t >= T) return;
    int ai = ii[t], aj = ij[t], ak = ik[t];
    float pix = pos[ai * 3], piy = pos[ai * 3 + 1], piz = pos[ai * 3 + 2];
    float jx = pos[aj * 3] - pix, jy = pos[aj * 3 + 1] - piy, jz = pos[aj * 3 + 2] - piz;
    float kx = pos[ak * 3] - pix, ky = pos[ak * 3 + 1] - piy, kz = pos[ak * 3 + 2] - piz;
    float a = jx * kx + jy * ky + jz * kz;
    float cx = jy * kz - jz * ky, cy = jz * kx - jx * kz, cz = jx * ky - jy * kx;
    float b = sqrtf(cx * cx + cy * cy + cz * cz);
    float r = sqrtf(a * a + b * b);
    float cang = (r > 0.f) ? a / r : 1.f;    // cos(atan2(b,a))
    const float I4P = 0.07957747154594767f;  // 1/(4*pi)
    float pm = 1.f, pc = cang;
    cbf[(size_t)t * 7 + 0] = sqrtf(1.f * I4P);
    cbf[(size_t)t * 7 + 1] = sqrtf(3.f * I4P) * pc;
    for (int l = 2; l < 7; l++) {
        float pn = ((float)(2 * l - 1) * cang * pc - (float)(l - 1) * pm) / (float)l;
        pm = pc; pc = pn;
        cbf[(size_t)t * 7 + l] = sqrtf((float)(2 * l + 1) * I4P) * pn;
    }
}

// sb[t,8] = (rad[idx_kj[t],:42] * cbf[t,l]) @ W_sbf[42,8]
__global__ __launch_bounds__(256) void sb_kernel(
    const float* __restrict__ rade, const float* __restrict__ cbf,
    const int* __restrict__ idx_kj, const float* __restrict__ Wsbf,
    float* __restrict__ sb, int T)
{
    __shared__ float Ws[42 * 8];
    for (int i = threadIdx.x; i < 336; i += 256) Ws[i] = Wsbf[i];
    __syncthreads();
    int t = blockIdx.x * 256 + threadIdx.x;
    if (t >= T) return;
    int kj = idx_kj[t];
    const float* rr = rade + (size_t)kj * 42;
    const float* cc = cbf + (size_t)t * 7;
    float o[8] = {0, 0, 0, 0, 0, 0, 0, 0};
    for (int l = 0; l < 7; l++) {
        float cl = cc[l];
        for (int r = 0; r < 6; r++) {
            float s = rr[l * 6 + r] * cl;
            const float* w = Ws + (l * 6 + r) * 8;
            #pragma unroll
            for (int q = 0; q < 8; q++) o[q] += s * w[q];
        }
    }
    #pragma unroll
    for (int q = 0; q < 8; q++) sb[(size_t)t * 8 + q] = o[q];
}

// One wave per output edge e': G[e',l,:] = sum_{t in seg(e')} sb[t,l]*xkj[kj(t),:]
// idx_ji is sorted -> binary-search segment, no atomics.
__global__ __launch_bounds__(256) void g_kernel(
    const float* __restrict__ xkj, const float* __restrict__ sb,
    const int* __restrict__ idx_kj, const int* __restrict__ idx_ji,
    int T, int e0, int e1, float* __restrict__ G)
{
    int wid = blockIdx.x * 8 + (threadIdx.x >> 5);
    int e = e0 + wid;
    if (e >= e1) return;
    int lane = threadIdx.x & 31;

    int lo = 0, hi = T;
    while (lo < hi) { int mid = (lo + hi) >> 1; if (idx_ji[mid] < e) lo = mid + 1; else hi = mid; }
    int start = lo;
    hi = T;
    while (lo < hi) { int mid = (lo + hi) >> 1; if (idx_ji[mid] <= e) lo = mid + 1; else hi = mid; }
    int end = lo;

    float g[8][4];
    #pragma unroll
    for (int l = 0; l < 8; l++) { g[l][0] = g[l][1] = g[l][2] = g[l][3] = 0.f; }

    for (int t = start; t < end; t++) {
        int kj = idx_kj[t];
        if (t + 1 < end)   // gfx1250: global_prefetch for next gathered row
            __builtin_prefetch(&xkj[(size_t)idx_kj[t + 1] * 128 + lane * 4], 0, 1);
        float4 xv = *(const float4*)(xkj + (size_t)kj * 128 + lane * 4);
        float4 s0 = *(const float4*)(sb + (size_t)t * 8);
        float4 s1 = *(const float4*)(sb + (size_t)t * 8 + 4);
        g[0][0] += s0.x * xv.x; g[0][1] += s0.x * xv.y; g[0][2] += s0.x * xv.z; g[0][3] += s0.x * xv.w;
        g[1][0] += s0.y * xv.x; g[1][1] += s0.y * xv.y; g[1][2] += s0.y * xv.z; g[1][3] += s0.y * xv.w;
        g[2][0] += s0.z * xv.x; g[2][1] += s0.z * xv.y; g[2][2] += s0.z * xv.z; g[2][3] += s0.z * xv.w;
        g[3][0] += s0.w * xv.x; g[3][1] += s0.w * xv.y; g[3][2] += s0.w * xv.z; g[3][3] += s0.w * xv.w;
        g[4][0] += s1.x * xv.x; g[4][1] += s1.x * xv.y; g[4][2] += s1.x * xv.z; g[4][3] += s1.x * xv.w;
        g[5][0] += s1.y * xv.x; g[5][1] += s1.y * xv.y; g[5][2] += s1.y * xv.z; g[5][3] += s1.y * xv.w;
        g[6][0] += s1.z * xv.x; g[6][1] += s1.z * xv.y; g[6][2] += s1.z * xv.z; g[6][3] += s1.z * xv.w;
        g[7][0] += s1.w * xv.x; g[7][1] += s1.w * xv.y; g[7][2] += s1.w * xv.z; g[7][3] += s1.w * xv.w;
    }
    float* Gr = G + (size_t)(e - e0) * 1024;
    #pragma unroll
    for (int l = 0; l < 8; l++) {
        float4 o = make_float4(g[l][0], g[l][1], g[l][2], g[l][3]);
        *(float4*)(Gr + l * 128 + lane * 4) = o;
    }
}

// WBT[l*128+j, i] = W_bil[i, l, j]  (so msg = G[E,1024] @ WBT[1024,128])
__global__ __launch_bounds__(256) void wbt_kernel(
    const float* __restrict__ Wbil, float* __restrict__ WBT)
{
    int idx = blockIdx.x * 256 + threadIdx.x;
    if (idx >= 128 * 8 * 128) return;
    int i  = idx >> 10;
    int l  = (idx >> 7) & 7;
    int jj = idx & 127;
    WBT[(size_t)(l * 128 + jj) * 128 + i] = Wbil[idx];
}

// A_embed[e] = [hx[i[e]] | hx[j[e]] | xr[e]]
__global__ __launch_bounds__(256) void concat_kernel(
    const float* __restrict__ hx, const float* __restrict__ xr,
    const int* __restrict__ ei, int E, float* __restrict__ Aemb)
{
    long idx = (long)blockIdx.x * 256 + threadIdx.x;
    if (idx >= (long)E * 96) return;
    int e = (int)(idx / 96), q4 = (int)(idx % 96);
    float4 v;
    if (q4 < 32)      v = ((const float4*)hx)[(size_t)ei[E + e] * 32 + q4];
    else if (q4 < 64) v = ((const float4*)hx)[(size_t)ei[e]     * 32 + (q4 - 32)];
    else              v = ((const float4*)xr)[(size_t)e         * 32 + (q4 - 64)];
    ((float4*)Aemb)[(size_t)e * 96 + q4] = v;
}

// Out-block front: tnode[i[e],c] += (rbf[e]@W_rbf)[c] * x[e,c]
__global__ __launch_bounds__(256) void outblock_scatter_kernel(
    const float* __restrict__ rbf, const float* __restrict__ Wrb,
    const float* __restrict__ x, const int* __restrict__ i_idx,
    float* __restrict__ tnode, int E)
{
    long idx = (long)blockIdx.x * 256 + threadIdx.x;
    if (idx >= (long)E * 128) return;
    int e = (int)(idx >> 7), col = (int)(idx & 127);
    float rb = 0.f;
    #pragma unroll
    for (int q = 0; q < 6; q++) rb += rbf[(size_t)e * 6 + q] * Wrb[q * 128 + col];
    atomicAdd(&tnode[(size_t)i_idx[e] * 128 + col], rb * x[(size_t)e * 128 + col]);
}

__global__ void colsum_kernel(const float* __restrict__ t, int M,
                              float* __restrict__ svec)
{
    int col = threadIdx.x;             // 128 threads
    int r0 = blockIdx.x * 256;
    int r1 = r0 + 256; if (r1 > M) r1 = M;
    float s = 0.f;
    for (int r = r0; r < r1; r++) s += t[(size_t)r * 128 + col];
    atomicAdd(&svec[col], s);
}

__global__ void outvec_kernel(const float* __restrict__ svec,
                              const float* __restrict__ Wout,
                              float* __restrict__ out)
{
    __shared__ float s[128];
    s[threadIdx.x] = svec[threadIdx.x];
    __syncthreads();
    float v = 0.f;
    for (int c = 0; c < 128; c++) v += s[c] * Wout[c * 128 + threadIdx.x];
    out[threadIdx.x] += v;
}

__global__ void zero_kernel(float* __restrict__ p, long n)
{
    long i = (long)blockIdx.x * 256 + threadIdx.x;
    if (i < n) p[i] = 0.f;
}

// ---------------------------------------------------------------------------

extern "C" void kernel_launch(void* const* d_in, const int* in_sizes, int n_in,
                              void* d_out, int out_size, void* d_ws, size_t ws_size,
                              hipStream_t stream)
{
    const float* node_feats = (const float*)d_in[0];
    const float* pos        = (const float*)d_in[1];
    const int*   edge_index = (const int*)d_in[2];
    const int*   idx_i      = (const int*)d_in[3];
    const int*   idx_j      = (const int*)d_in[4];
    const int*   idx_k      = (const int*)d_in[5];
    const int*   idx_kj     = (const int*)d_in[6];
    const int*   idx_ji     = (const int*)d_in[7];
    const float* sbf_zeros  = (const float*)d_in[8];
    const float* sbf_norms  = (const float*)d_in[9];
    auto P = [&](int k) { return (const float*)d_in[k]; };
    // params pytree flatten (sorted keys), leaves start at 10:
    //  10 emb.W[384,128] 11 emb.W_rbf[6,128] 12 emb.W_x[64,128]
    //  13 emb.b 14 emb.b_rbf 15 emb.b_x
    //  16 + bi*21: int block {W_bil,W_ji,W_kj,W_rbf,W_sbf,W_skip,
    //    a0W1,a0W2,a0b1,a0b2, a1W1,a1W2,a1b1,a1b2, r0W1,r0W2,r0b1,r0b2,
    //    b_ji,b_kj,b_skip}
    //  100 + ob*8: out block {W_out,W_rbf, l0W,l0b, l1W,l1b, l2W,l2b}
    //  140: rbf_freq (unused; frequencies are (r+1)*pi)

    const int Nn = in_sizes[0] / 64;
    const int E  = in_sizes[2] / 2;
    const int T  = in_sizes[3];
    const int GCH = 16384;                 // G-buffer edge chunk

    size_t off = 0;
    auto alloc = [&](size_t bytes) -> float* {
        float* p = (float*)((char*)d_ws + off);
        off = (off + bytes + 255) & ~(size_t)255;
        return p;
    };
    float* RBF  = alloc((size_t)E * 6 * 4);
    float* RADE = alloc((size_t)E * 42 * 4);
    float* CBF  = alloc((size_t)T * 7 * 4);
    float* HX   = alloc((size_t)Nn * 128 * 4);
    float* XR   = alloc((size_t)E * 128 * 4);
    float* X0   = alloc((size_t)E * 128 * 4);
    float* X1   = alloc((size_t)E * 128 * 4);
    float* XJI  = alloc((size_t)E * 128 * 4);
    float* XKJ  = alloc((size_t)E * 128 * 4);
    float* HB   = alloc((size_t)E * 128 * 4);
    float* U    = alloc((size_t)E * 128 * 4);
    float* SB   = alloc((size_t)T * 8 * 4);
    float* GBUF = alloc((size_t)GCH * 1024 * 4);
    float* WBT  = alloc((size_t)1024 * 128 * 4);
    float* TN0  = alloc((size_t)Nn * 128 * 4);
    float* TN1  = alloc((size_t)Nn * 128 * 4);
    float* SVEC = alloc(128 * 4);
    float* AEMB = XJI;   // [E,384] aliases XJI+XKJ+HB (unused during embedding)

    auto gemm = [&](const float* A, int lda, const float* B, const float* bias,
                    const float* post, const float* rbf6, const float* Wr,
                    float* C, int M, int K, int pm) {
        wmma_gemm_n128<<<dim3((M + 127) / 128), dim3(256), 0, stream>>>(
            A, lda, B, bias, post, rbf6, Wr, C, M, K, pm);
    };
    auto zero = [&](float* p, long n) {
        zero_kernel<<<(unsigned)((n + 255) / 256), 256, 0, stream>>>(p, n);
    };

    auto run_out_block = [&](int ob, const float* x) {
        int b = 100 + ob * 8;
        zero(TN0, (long)Nn * 128);
        outblock_scatter_kernel<<<(unsigned)(((long)E * 128 + 255) / 256), 256, 0, stream>>>(
            RBF, P(b + 1), x, edge_index + E, TN0, E);
        gemm(TN0, 128, P(b + 2), P(b + 3), nullptr, nullptr, nullptr, TN1, Nn, 128, PM_SILU);
        gemm(TN1, 128, P(b + 4), P(b + 5), nullptr, nullptr, nullptr, TN0, Nn, 128, PM_SILU);
        gemm(TN0, 128, P(b + 6), P(b + 7), nullptr, nullptr, nullptr, TN1, Nn, 128, PM_SILU);
        zero(SVEC, 128);
        colsum_kernel<<<(Nn + 255) / 256, 128, 0, stream>>>(TN1, Nn, SVEC);
        outvec_kernel<<<1, 128, 0, stream>>>(SVEC, P(b + 0), (float*)d_out);
    };

    // ---------------- pipeline ----------------
    zero((float*)d_out, 128);

    edge_geom_kernel<<<(E + 255) / 256, 256, 0, stream>>>(
        pos, edge_index, E, sbf_zeros, sbf_norms, RBF, RADE);
    trip_geom_kernel<<<(T + 255) / 256, 256, 0, stream>>>(
        pos, idx_i, idx_j, idx_k, CBF, T);

    // Embedding
    gemm(node_feats, 64, P(12), P(15), nullptr, nullptr, nullptr, HX, Nn, 64, PM_NONE);
    gemm(RBF, 6, P(11), P(14), nullptr, nullptr, nullptr, XR, E, 6, PM_SILU);
    concat_kernel<<<(unsigned)(((long)E * 96 + 255) / 256), 256, 0, stream>>>(
        HX, XR, edge_index, E, AEMB);
    gemm(AEMB, 384, P(10), P(13), nullptr, nullptr, nullptr, X0, E, 384, PM_SILU);

    float* x_cur = X0;
    float* x_alt = X1;
    run_out_block(0, x_cur);

    for (int bi = 0; bi < 4; bi++) {
        int b = 16 + bi * 21;
        const float *W_bil = P(b + 0), *W_ji = P(b + 1), *W_kj = P(b + 2);
        const float *W_rbf = P(b + 3), *W_sbf = P(b + 4), *W_skip = P(b + 5);
        const float *a0W1 = P(b + 6),  *a0W2 = P(b + 7),  *a0b1 = P(b + 8),  *a0b2 = P(b + 9);
        const float *a1W1 = P(b + 10), *a1W2 = P(b + 11), *a1b1 = P(b + 12), *a1b2 = P(b + 13);
        const float *r0W1 = P(b + 14), *r0W2 = P(b + 15), *r0b1 = P(b + 16), *r0b2 = P(b + 17);
        const float *b_ji = P(b + 18), *b_kj = P(b + 19), *b_skip = P(b + 20);

        gemm(x_cur, 128, W_ji, b_ji, nullptr, nullptr, nullptr, XJI, E, 128, PM_SILU);
        gemm(x_cur, 128, W_kj, b_kj, nullptr, RBF, W_rbf, XKJ, E, 128, PM_SILU_RBMUL);
        sb_kernel<<<(T + 255) / 256, 256, 0, stream>>>(RADE, CBF, idx_kj, W_sbf, SB, T);
        wbt_kernel<<<(128 * 8 * 128 + 255) / 256, 256, 0, stream>>>(W_bil, WBT);

        for (int e0 = 0; e0 < E; e0 += GCH) {
            int cm = E - e0; if (cm > GCH) cm = GCH;
            g_kernel<<<(cm + 7) / 8, 256, 0, stream>>>(
                XKJ, SB, idx_kj, idx_ji, T, e0, e0 + cm, GBUF);
            gemm(GBUF, 1024, WBT, nullptr, XJI + (size_t)e0 * 128, nullptr, nullptr,
                 HB + (size_t)e0 * 128, cm, 1024, PM_ADD);     // h = x_ji + msg
        }
        // before-residual
        gemm(HB, 128, r0W1, r0b1, nullptr, nullptr, nullptr, U, E, 128, PM_SILU);
        gemm(U, 128, r0W2, r0b2, HB, nullptr, nullptr, HB, E, 128, PM_SILU_ADD);
        // skip: x_new = silu(h@W_skip+b)+x
        gemm(HB, 128, W_skip, b_skip, x_cur, nullptr, nullptr, x_alt, E, 128, PM_SILU_ADD);
        // after-residuals x2
        gemm(x_alt, 128, a0W1, a0b1, nullptr, nullptr, nullptr, U, E, 128, PM_SILU);
        gemm(U, 128, a0W2, a0b2, x_alt, nullptr, nullptr, x_alt, E, 128, PM_SILU_ADD);
        gemm(x_alt, 128, a1W1, a1b1, nullptr, nullptr, nullptr, U, E, 128, PM_SILU);
        gemm(U, 128, a1W2, a1b2, x_alt, nullptr, nullptr, x_alt, E, 128, PM_SILU_ADD);

        float* tmp = x_cur; x_cur = x_alt; x_alt = tmp;
        run_out_block(bi + 1, x_cur);
    }
    (void)n_in; (void)out_size; (void)ws_size;
}